// GATv2_pr_15796889715060
// MI455X (gfx1250) — compile-verified
//
#include <hip/hip_runtime.h>
#include <hip/hip_bf16.h>

typedef float v2f __attribute__((ext_vector_type(2)));
typedef float v8f __attribute__((ext_vector_type(8)));

#define NEG_SLOPE 0.2f
#define BN_EPS 1e-5f

// ---------------------------------------------------------------------------
// f32 WMMA GEMM: C[M,Nc] = relu?(A[M,K] @ W[K,Nc] + bias)
// One wave (32 threads) per 16x64 output strip: 4 accumulators share one
// A fragment per K-step -> 4 back-to-back V_WMMA_F32_16X16X4_F32 per 10 loads.
// Requires Nc % 64 == 0 (true for all call sites: 256, 64, 64, 64).
// EXEC stays all-ones (no divergent control flow before the WMMAs).
// ---------------------------------------------------------------------------
__global__ __launch_bounds__(32)
void wmma_gemm_f32(const float* __restrict__ A, const float* __restrict__ W,
                   const float* __restrict__ bias, float* __restrict__ C,
                   int M, int K, int Nc, int relu)
{
    const int lane = threadIdx.x;        // 0..31
    const int half = lane >> 4;          // 0: lanes 0-15, 1: lanes 16-31
    const int l    = lane & 15;
    const int m0   = blockIdx.y * 16;
    const int n0   = blockIdx.x * 64;

    // Clamp row index so out-of-range tiles still load valid memory
    // (no divergent branches -> EXEC remains all ones for the WMMA).
    int arow = m0 + l;
    arow = arow < M ? arow : (M - 1);
    const float* __restrict__ Arow = A + (size_t)arow * K;

    v8f acc0 = {}, acc1 = {}, acc2 = {}, acc3 = {};
    for (int kk = 0; kk < K; kk += 4) {
        const int ka = kk + half * 2;                 // this half-wave's K pair
        // A fragment: 16x4 f32, lane holds A[row][ka], A[row][ka+1]
        v2f a;
        a.x = Arow[ka];
        a.y = Arow[ka + 1];
        // B fragments for 4 adjacent 16-col tiles (rows ka, ka+1 of W)
        const float* __restrict__ Wr0 = W + (size_t)ka * Nc + n0 + l;
        const float* __restrict__ Wr1 = W + (size_t)(ka + 1) * Nc + n0 + l;
        v2f b0, b1, b2, b3;
        b0.x = Wr0[0];  b0.y = Wr1[0];
        b1.x = Wr0[16]; b1.y = Wr1[16];
        b2.x = Wr0[32]; b2.y = Wr1[32];
        b3.x = Wr0[48]; b3.y = Wr1[48];
        acc0 = __builtin_amdgcn_wmma_f32_16x16x4_f32(
            false, a, false, b0, (short)0, acc0, false, false);
        acc1 = __builtin_amdgcn_wmma_f32_16x16x4_f32(
            false, a, false, b1, (short)0, acc1, false, false);
        acc2 = __builtin_amdgcn_wmma_f32_16x16x4_f32(
            false, a, false, b2, (short)0, acc2, false, false);
        acc3 = __builtin_amdgcn_wmma_f32_16x16x4_f32(
            false, a, false, b3, (short)0, acc3, false, false);
    }

    // Store: D layout -> VGPR r holds row m0 + r + 8*half, col = tile_col + l
    v8f accs[4] = {acc0, acc1, acc2, acc3};
#pragma unroll
    for (int t = 0; t < 4; ++t) {
        const int col = n0 + t * 16 + l;
        const float bv = bias[col];
#pragma unroll
        for (int r = 0; r < 8; ++r) {
            const int row = m0 + r + half * 8;
            if (row < M) {
                float v = accs[t][r] + bv;
                if (relu) v = fmaxf(v, 0.0f);
                C[(size_t)row * Nc + col] = v;
            }
        }
    }
}

// ---------------------------------------------------------------------------
// Edge scoring: score[e,h] = sum_c att[h,c]*leaky_relu(xl[src,h,c]+xr[dst,h,c])
// ---------------------------------------------------------------------------
__global__ void edge_score_kernel(const float* __restrict__ xl,
                                  const float* __restrict__ xr,
                                  const int* __restrict__ src,
                                  const int* __restrict__ dst,
                                  const float* __restrict__ att,
                                  float* __restrict__ score,
                                  int E, int heads, int ch)
{
    int gid = blockIdx.x * blockDim.x + threadIdx.x;
    if (gid >= E * heads) return;
    int e = gid / heads;
    int h = gid - e * heads;
    const float* pl = xl + (size_t)src[e] * heads * ch + h * ch;
    const float* pr = xr + (size_t)dst[e] * heads * ch + h * ch;
    const float* pa = att + h * ch;
    float sc = 0.0f;
    for (int c = 0; c < ch; ++c) {
        float v = pl[c] + pr[c];
        v = v > 0.0f ? v : v * NEG_SLOPE;
        sc = fmaf(v, pa[c], sc);
    }
    score[gid] = sc;
}

// Ordered-uint encoding for float atomicMax (works for all finite floats).
__device__ __forceinline__ unsigned flipf(float f) {
    unsigned u = __float_as_uint(f);
    return (u & 0x80000000u) ? ~u : (u | 0x80000000u);
}
__device__ __forceinline__ float unflipf(unsigned u) {
    return (u & 0x80000000u) ? __uint_as_float(u ^ 0x80000000u)
                             : __uint_as_float(~u);
}

__global__ void seg_max_kernel(const float* __restrict__ score,
                               const int* __restrict__ dst,
                               unsigned* __restrict__ smax,
                               int E, int heads)
{
    int gid = blockIdx.x * blockDim.x + threadIdx.x;
    if (gid >= E * heads) return;
    int e = gid / heads;
    int h = gid - e * heads;
    atomicMax(&smax[(size_t)dst[e] * heads + h], flipf(score[gid]));
}

__global__ void edge_exp_sum_kernel(float* __restrict__ score,
                                    const int* __restrict__ dst,
                                    const unsigned* __restrict__ smax,
                                    float* __restrict__ ssum,
                                    int E, int heads)
{
    int gid = blockIdx.x * blockDim.x + threadIdx.x;
    if (gid >= E * heads) return;
    int e = gid / heads;
    int h = gid - e * heads;
    size_t di = (size_t)dst[e] * heads + h;
    unsigned menc = smax[di];
    float m = (menc == 0u) ? 0.0f : unflipf(menc);   // sentinel == -inf -> 0
    float ex = expf(score[gid] - m);
    score[gid] = ex;
    atomicAdd(&ssum[di], ex);
}

__global__ void edge_alpha_kernel(float* __restrict__ score,
                                  const int* __restrict__ dst,
                                  const float* __restrict__ ssum,
                                  int E, int heads)
{
    int gid = blockIdx.x * blockDim.x + threadIdx.x;
    if (gid >= E * heads) return;
    int e = gid / heads;
    int h = gid - e * heads;
    score[gid] = score[gid] / (ssum[(size_t)dst[e] * heads + h] + 1e-16f);
}

__global__ void edge_aggregate_kernel(const float* __restrict__ xl,
                                      const float* __restrict__ alpha,
                                      const int* __restrict__ src,
                                      const int* __restrict__ dst,
                                      float* __restrict__ acc,
                                      int E, int heads, int ch)
{
    long gid = (long)blockIdx.x * blockDim.x + threadIdx.x;
    long tot = (long)E * heads * ch;
    if (gid >= tot) return;
    int c = (int)(gid % ch);
    long t = gid / ch;
    int h = (int)(t % heads);
    int e = (int)(t / heads);
    float a = alpha[(size_t)e * heads + h];
    float v = a * xl[(size_t)src[e] * heads * ch + h * ch + c];
    atomicAdd(&acc[(size_t)dst[e] * heads * ch + h * ch + c], v);
}

__global__ void bias_relu_kernel(float* __restrict__ a,
                                 const float* __restrict__ bias,
                                 long total, int Nc)
{
    long gid = (long)blockIdx.x * blockDim.x + threadIdx.x;
    if (gid >= total) return;
    int j = (int)(gid % Nc);
    a[gid] = fmaxf(a[gid] + bias[j], 0.0f);
}

__global__ void pool_sum_kernel(const float* __restrict__ h,
                                const int* __restrict__ batch,
                                float* __restrict__ pooled,
                                float* __restrict__ cnt,
                                int Nn, int F)
{
    long gid = (long)blockIdx.x * blockDim.x + threadIdx.x;
    long tot = (long)Nn * F;
    if (gid >= tot) return;
    int n = (int)(gid / F);
    int j = (int)(gid % F);
    atomicAdd(&pooled[(size_t)batch[n] * F + j], h[gid]);
    if (j == 0) atomicAdd(&cnt[batch[n]], 1.0f);
}

__global__ void pool_div_kernel(float* __restrict__ pooled,
                                const float* __restrict__ cnt,
                                int Bb, int F)
{
    int gid = blockIdx.x * blockDim.x + threadIdx.x;
    if (gid >= Bb * F) return;
    pooled[gid] = pooled[gid] / fmaxf(cnt[gid / F], 1.0f);
}

__global__ void fuse_bn_kernel(const float* __restrict__ pooled,
                               const float* __restrict__ doc_emb,
                               const float* __restrict__ gamma,
                               const float* __restrict__ beta,
                               const float* __restrict__ mean,
                               const float* __restrict__ var,
                               float* __restrict__ fusion,
                               int Bb, int Hc)
{
    int F2 = 2 * Hc;
    int gid = blockIdx.x * blockDim.x + threadIdx.x;
    if (gid >= Bb * F2) return;
    int b = gid / F2;
    int j = gid - b * F2;
    float v = (j < Hc) ? pooled[(size_t)b * Hc + j]
                       : doc_emb[(size_t)b * Hc + (j - Hc)];
    fusion[gid] = (v - mean[j]) * rsqrtf(var[j] + BN_EPS) * gamma[j] + beta[j];
}

__global__ void heads_out_kernel(const float* __restrict__ fused,
                                 const float* __restrict__ taskW,
                                 const float* __restrict__ taskb,
                                 const float* __restrict__ timeW,
                                 const float* __restrict__ timeb,
                                 float* __restrict__ out,
                                 int Bb, int Hc, int OUT)
{
    int gid = blockIdx.x * blockDim.x + threadIdx.x;
    if (gid >= Bb * (OUT + 1)) return;
    int b = gid / (OUT + 1);
    int o = gid - b * (OUT + 1);
    const float* f = fused + (size_t)b * Hc;
    if (o < OUT) {
        float s = taskb[o];
        for (int c = 0; c < Hc; ++c) s = fmaf(f[c], taskW[c * OUT + o], s);
        out[(size_t)b * OUT + o] = s;                  // task_output [B,OUT]
    } else {
        float s = timeb[0];
        for (int c = 0; c < Hc; ++c) s = fmaf(f[c], timeW[c], s);
        out[(size_t)Bb * OUT + b] = s;                 // time_output [B,1]
    }
}

// ---------------------------------------------------------------------------
extern "C" void kernel_launch(void* const* d_in, const int* in_sizes, int n_in,
                              void* d_out, int out_size, void* d_ws, size_t ws_size,
                              hipStream_t stream)
{
    const float* x     = (const float*)d_in[0];
    const int*   src   = (const int*)d_in[1];
    const int*   dst   = (const int*)d_in[2];
    const int*   batch = (const int*)d_in[3];
    const float* doc   = (const float*)d_in[4];
    const float* Wl1   = (const float*)d_in[5];
    const float* bl1   = (const float*)d_in[6];
    const float* Wr1   = (const float*)d_in[7];
    const float* br1   = (const float*)d_in[8];
    const float* att1  = (const float*)d_in[9];
    const float* bias1 = (const float*)d_in[10];
    const float* Wl2   = (const float*)d_in[11];
    const float* bl2   = (const float*)d_in[12];
    const float* Wr2   = (const float*)d_in[13];
    const float* br2   = (const float*)d_in[14];
    const float* att2  = (const float*)d_in[15];
    const float* bias2 = (const float*)d_in[16];
    const float* docW  = (const float*)d_in[17];
    const float* docb  = (const float*)d_in[18];
    const float* bng   = (const float*)d_in[19];
    const float* bnb   = (const float*)d_in[20];
    const float* bnm   = (const float*)d_in[21];
    const float* bnv   = (const float*)d_in[22];
    const float* fusW  = (const float*)d_in[23];
    const float* fusb  = (const float*)d_in[24];
    const float* taskW = (const float*)d_in[25];
    const float* taskb = (const float*)d_in[26];
    const float* timeW = (const float*)d_in[27];
    const float* timeb = (const float*)d_in[28];

    const int IN = 128, Hc = 64, HEADS = 4, DOC = 256, OUT = 16;
    const int Nn = in_sizes[0] / IN;      // 50000
    const int Ee = in_sizes[1];           // 800000
    const int Bb = in_sizes[4] / DOC;     // 64
    const int F1 = HEADS * Hc;            // 256

    // ---- workspace layout (floats) ----
    float* ws = (float*)d_ws;
    size_t off = 0;
    float*    xl1   = ws + off; off += (size_t)Nn * F1;
    float*    xr1   = ws + off; off += (size_t)Nn * F1;
    float*    acc1  = ws + off; off += (size_t)Nn * F1;   // becomes h1 in place
    float*    score = ws + off; off += (size_t)Ee * HEADS;
    unsigned* smax  = (unsigned*)(ws + off); off += (size_t)Nn * HEADS;
    float*    ssum  = ws + off; off += (size_t)Nn * HEADS;
    float*    pooled= ws + off; off += (size_t)Bb * Hc;
    float*    cnt   = ws + off; off += (size_t)Bb;
    float*    docE  = ws + off; off += (size_t)Bb * Hc;
    float*    fusion= ws + off; off += (size_t)Bb * 2 * Hc;
    float*    fused = ws + off; off += (size_t)Bb * Hc;
    // layer-2 aliases (layer-1 regions are done being read when these are written)
    float*    xl2   = xl1;                       // [Nn,64] in first part of xl1
    float*    acc2  = xl1 + (size_t)Nn * Hc;     // distinct range inside xl1 region
    float*    xr2   = xr1;
    float*    score2= score;
    unsigned* smax2 = smax;
    float*    ssum2 = ssum;

    const int TB = 256;
    auto nblk = [](long n, int b) { return (unsigned)((n + b - 1) / b); };

    // ---- init ----
    hipMemsetAsync(acc1, 0, (size_t)Nn * F1 * 4, stream);
    hipMemsetAsync(smax, 0, (size_t)Nn * HEADS * 4, stream);
    hipMemsetAsync(ssum, 0, (size_t)Nn * HEADS * 4, stream);
    hipMemsetAsync(pooled, 0,
                   ((size_t)Bb * Hc + Bb + (size_t)Bb * Hc + (size_t)Bb * 2 * Hc +
                    (size_t)Bb * Hc) * 4, stream);

    // ---- GAT layer 1 ----
    {
        dim3 g(F1 / 64, (Nn + 15) / 16);
        wmma_gemm_f32<<<g, 32, 0, stream>>>(x, Wl1, bl1, xl1, Nn, IN, F1, 0);
        wmma_gemm_f32<<<g, 32, 0, stream>>>(x, Wr1, br1, xr1, Nn, IN, F1, 0);
    }
    edge_score_kernel<<<nblk((long)Ee * HEADS, TB), TB, 0, stream>>>(
        xl1, xr1, src, dst, att1, score, Ee, HEADS, Hc);
    seg_max_kernel<<<nblk((long)Ee * HEADS, TB), TB, 0, stream>>>(
        score, dst, smax, Ee, HEADS);
    edge_exp_sum_kernel<<<nblk((long)Ee * HEADS, TB), TB, 0, stream>>>(
        score, dst, smax, ssum, Ee, HEADS);
    edge_alpha_kernel<<<nblk((long)Ee * HEADS, TB), TB, 0, stream>>>(
        score, dst, ssum, Ee, HEADS);
    edge_aggregate_kernel<<<nblk((long)Ee * HEADS * Hc, TB), TB, 0, stream>>>(
        xl1, score, src, dst, acc1, Ee, HEADS, Hc);
    bias_relu_kernel<<<nblk((long)Nn * F1, TB), TB, 0, stream>>>(
        acc1, bias1, (long)Nn * F1, F1);            // acc1 -> h1

    // ---- GAT layer 2 (1 head) ----
    hipMemsetAsync(acc2, 0, (size_t)Nn * Hc * 4, stream);
    hipMemsetAsync(smax2, 0, (size_t)Nn * 4, stream);
    hipMemsetAsync(ssum2, 0, (size_t)Nn * 4, stream);
    {
        dim3 g(Hc / 64, (Nn + 15) / 16);
        wmma_gemm_f32<<<g, 32, 0, stream>>>(acc1, Wl2, bl2, xl2, Nn, F1, Hc, 0);
        wmma_gemm_f32<<<g, 32, 0, stream>>>(acc1, Wr2, br2, xr2, Nn, F1, Hc, 0);
    }
    edge_score_kernel<<<nblk((long)Ee, TB), TB, 0, stream>>>(
        xl2, xr2, src, dst, att2, score2, Ee, 1, Hc);
    seg_max_kernel<<<nblk((long)Ee, TB), TB, 0, stream>>>(
        score2, dst, smax2, Ee, 1);
    edge_exp_sum_kernel<<<nblk((long)Ee, TB), TB, 0, stream>>>(
        score2, dst, smax2, ssum2, Ee, 1);
    edge_alpha_kernel<<<nblk((long)Ee, TB), TB, 0, stream>>>(
        score2, dst, ssum2, Ee, 1);
    edge_aggregate_kernel<<<nblk((long)Ee * Hc, TB), TB, 0, stream>>>(
        xl2, score2, src, dst, acc2, Ee, 1, Hc);
    bias_relu_kernel<<<nblk((long)Nn * Hc, TB), TB, 0, stream>>>(
        acc2, bias2, (long)Nn * Hc, Hc);            // acc2 -> h2

    // ---- pooling + doc + fusion + heads ----
    pool_sum_kernel<<<nblk((long)Nn * Hc, TB), TB, 0, stream>>>(
        acc2, batch, pooled, cnt, Nn, Hc);
    pool_div_kernel<<<nblk((long)Bb * Hc, TB), TB, 0, stream>>>(
        pooled, cnt, Bb, Hc);
    {
        dim3 g(Hc / 64, (Bb + 15) / 16);
        wmma_gemm_f32<<<g, 32, 0, stream>>>(doc, docW, docb, docE, Bb, DOC, Hc, 1);
    }
    fuse_bn_kernel<<<nblk((long)Bb * 2 * Hc, TB), TB, 0, stream>>>(
        pooled, docE, bng, bnb, bnm, bnv, fusion, Bb, Hc);
    {
        dim3 g(Hc / 64, (Bb + 15) / 16);
        wmma_gemm_f32<<<g, 32, 0, stream>>>(fusion, fusW, fusb, fused, Bb, 2 * Hc, Hc, 1);
    }
    heads_out_kernel<<<nblk((long)Bb * (OUT + 1), TB), TB, 0, stream>>>(
        fused, taskW, taskb, timeW, timeb, (float*)d_out, Bb, Hc, OUT);
}